// oampnet_base_37924561224367
// MI455X (gfx1250) — compile-verified
//
#include <hip/hip_runtime.h>

typedef __attribute__((ext_vector_type(2))) float v2f;
typedef __attribute__((ext_vector_type(8))) float v8f;

#define HS 33   // padded stride for 64x32 H and R/X tiles in LDS (bank-conflict-free)
#define MS 65   // padded stride for 64x64 M in LDS

__launch_bounds__(256)
__global__ void oampnet_cdna5_kernel(const float* __restrict__ Hg,
                                     const float* __restrict__ yg,
                                     const float* __restrict__ xg,
                                     const float* __restrict__ nsg,
                                     const float* __restrict__ thetag,
                                     const float* __restrict__ gammag,
                                     const float* __restrict__ csg,
                                     float* __restrict__ outg) {
  __shared__ float sM[64 * MS];   // v2t*H*H^T + lam*I, then Gauss-Jordan workspace
  __shared__ float sH[64 * HS];   // H
  __shared__ float sR[64 * HS];   // RHS copy of H -> becomes X = inv(M)*H
  __shared__ float sRT[64];       // rt
  __shared__ float sZ[32];        // zt accumulator (init x_out)
  __shared__ float sXO[32];
  __shared__ float sCst[8];
  __shared__ float sAcc[8];       // 0:||H||^2 1:||rt||^2 2:tr(H.X) 3:||X||^2 4:||c_t||^2
  __shared__ float sS[8];         // 0:v2t 1:lam 2:cc 3:inv(2*tau2)

  const int tid = threadIdx.x;
  const int b = blockIdx.x;
  const float* Hb = Hg + (size_t)b * 64 * 32;
  const float sig  = nsg[b];
  const float sig2 = sig * sig;
  const float theta = thetag[0];
  const float gamma = gammag[0];

  if (tid < 32) { float xv = xg[(size_t)b * 32 + tid]; sXO[tid] = xv; sZ[tid] = xv; }
  if (tid < 8)  { sCst[tid] = csg[tid]; sAcc[tid] = 0.0f; }

  // ---- async global->LDS fill of H into both sH and sR (CDNA5 async path,
  //      tracked by ASYNCcnt; B32 needs only 4B alignment so stride-33 is fine)
  #pragma unroll
  for (int t = 0; t < 8; ++t) {
    int e = tid + t * 256;
    int r = e >> 5, c = e & 31;
    unsigned int dH = (unsigned int)(uintptr_t)(&sH[r * HS + c]); // low 32b = LDS offset
    unsigned int dR = (unsigned int)(uintptr_t)(&sR[r * HS + c]);
    const float* src = Hb + e;
    asm volatile("global_load_async_to_lds_b32 %0, %2, off\n\t"
                 "global_load_async_to_lds_b32 %1, %2, off"
                 :: "v"(dH), "v"(dR), "v"(src) : "memory");
  }
  asm volatile("s_wait_asynccnt 0" ::: "memory");
  __syncthreads();

  // ---- ||H||_F^2 partial from LDS
  float hf = 0.0f;
  #pragma unroll
  for (int t = 0; t < 8; ++t) {
    int e = tid + t * 256;
    int r = e >> 5, c = e & 31;
    float v = sH[r * HS + c];
    hf += v * v;
  }
  atomicAdd(&sAcc[0], hf);
  if (tid < 64) {                       // rt = y - H x
    float s = 0.0f;
    #pragma unroll
    for (int j = 0; j < 32; ++j) s += sH[tid * HS + j] * sXO[j];
    float r = yg[(size_t)b * 64 + tid] - s;
    sRT[tid] = r;
    atomicAdd(&sAcc[1], r * r);
  }
  __syncthreads();
  if (tid == 0) {
    float v2t = (sAcc[1] - 32.0f * sig2) / sAcc[0];   // NR = 32
    sS[0] = fmaxf(v2t, 1e-9f);
    sS[1] = 0.5f * sig2;
  }
  __syncthreads();

  const int lane = tid & 31;
  const int wave = tid >> 5;
  const int lr = lane & 15;            // fragment row/col within tile
  const int kh = (lane >> 4) << 1;     // K offset 0 or 2 (upper lane half)

  // ---- M = v2t * H*H^T + lam*I via V_WMMA_F32_16X16X4_F32 (16 tiles, 2/wave)
  {
    const float v2t = sS[0], lam = sS[1];
    #pragma unroll
    for (int tt = 0; tt < 2; ++tt) {
      int t = wave * 2 + tt;
      int I = t >> 2, J = t & 3;
      v8f acc = {};
      #pragma unroll
      for (int s = 0; s < 8; ++s) {
        int k0 = 4 * s + kh;
        v2f a, bb;
        a[0]  = sH[(16 * I + lr) * HS + k0];
        a[1]  = sH[(16 * I + lr) * HS + k0 + 1];
        bb[0] = sH[(16 * J + lr) * HS + k0];
        bb[1] = sH[(16 * J + lr) * HS + k0 + 1];
        acc = __builtin_amdgcn_wmma_f32_16x16x4_f32(false, a, false, bb,
                                                    (short)0, acc, false, false);
      }
      #pragma unroll
      for (int r = 0; r < 8; ++r) {
        int gi = 16 * I + ((lane >> 4) << 3) + r;
        int gj = 16 * J + lr;
        sM[gi * MS + gj] = v2t * acc[r] + (gi == gj ? lam : 0.0f);
      }
    }
  }
  __syncthreads();

  // ---- Gauss-Jordan (SPD, no pivoting): M X = H, X left in sR (scaled at end)
  const int gi = tid & 63;   // row
  const int cg = tid >> 6;   // column group 0..3
  for (int k = 0; k < 64; ++k) {
    __syncthreads();
    if (gi != k) {
      float pivinv = 1.0f / sM[k * MS + k];
      float f = sM[gi * MS + k] * pivinv;
      for (int j = k + 1 + cg; j < 64; j += 4)
        sM[gi * MS + j] -= f * sM[k * MS + j];
      #pragma unroll
      for (int j = cg; j < 32; j += 4)
        sR[gi * HS + j] -= f * sR[k * HS + j];
    }
  }
  __syncthreads();
  {
    float dinv = 1.0f / sM[gi * MS + gi];
    #pragma unroll
    for (int j = cg; j < 32; j += 4) sR[gi * HS + j] *= dinv;
  }
  __syncthreads();

  // ---- tr(H.X) and ||X||^2 ; cc = 2*NT*gamma/tr = 32*gamma/tr (v2t cancels)
  {
    float t1 = 0.0f, t2 = 0.0f;
    #pragma unroll
    for (int t = 0; t < 8; ++t) {
      int e = tid + t * 256;
      int r = e >> 5, c = e & 31;
      float xv = sR[r * HS + c];
      t1 += sH[r * HS + c] * xv;
      t2 += xv * xv;
    }
    atomicAdd(&sAcc[2], t1);
    atomicAdd(&sAcc[3], t2);
  }
  __syncthreads();
  if (tid == 0) sS[2] = 32.0f * gamma / sAcc[2];
  __syncthreads();
  const float cc = sS[2];

  // ---- ||I - theta*cc*X^T H||_F^2 via WMMA (4 tiles on waves 0..3, K=64)
  float ctf = 0.0f;
  if (wave < 4) {                       // uniform per wave -> EXEC all ones inside
    int tm = (wave >> 1) & 1, tn = wave & 1;
    v8f acc = {};
    #pragma unroll
    for (int s = 0; s < 16; ++s) {
      int k0 = 4 * s + kh;
      v2f a, bb;
      a[0]  = sR[(k0)     * HS + 16 * tm + lr];   // A[m][k] = X[k][m]
      a[1]  = sR[(k0 + 1) * HS + 16 * tm + lr];
      bb[0] = sH[(k0)     * HS + 16 * tn + lr];   // B[k][n] = H[k][n]
      bb[1] = sH[(k0 + 1) * HS + 16 * tn + lr];
      acc = __builtin_amdgcn_wmma_f32_16x16x4_f32(false, a, false, bb,
                                                  (short)0, acc, false, false);
    }
    #pragma unroll
    for (int r = 0; r < 8; ++r) {
      int gi2 = 16 * tm + ((lane >> 4) << 3) + r;
      int gj2 = 16 * tn + lr;
      float ct = (gi2 == gj2 ? 1.0f : 0.0f) - theta * cc * acc[r];
      ctf += ct * ct;
    }
  }
  atomicAdd(&sAcc[4], ctf);

  // ---- zt = x_out + cc * X^T rt
  {
    int m = tid & 31, p = tid >> 5;
    float s = 0.0f;
    #pragma unroll
    for (int q = 0; q < 8; ++q) {
      int k = p * 8 + q;
      s += sR[k * HS + m] * sRT[k];
    }
    atomicAdd(&sZ[m], cc * s);
  }
  __syncthreads();
  if (tid == 0) {
    float tau2 = (sAcc[4] * sS[0] +
                  theta * theta * sig2 * cc * cc * sAcc[3] * 0.5f) / 32.0f;
    sS[3] = 1.0f / (2.0f * tau2);
  }
  __syncthreads();

  // ---- soft demap over 8-point constellation
  if (tid < 32) {
    float z = sZ[tid];
    float it = sS[3];
    float l[8], mx = -3.0e38f;
    #pragma unroll
    for (int k = 0; k < 8; ++k) {
      float d = z - sCst[k];
      l[k] = -d * d * it;
      mx = fmaxf(mx, l[k]);
    }
    float se = 0.0f, sw = 0.0f;
    #pragma unroll
    for (int k = 0; k < 8; ++k) {
      float e = __expf(l[k] - mx);
      se += e;
      sw += e * sCst[k];
    }
    outg[(size_t)b * 32 + tid] = sw / se;
  }
}

extern "C" void kernel_launch(void* const* d_in, const int* in_sizes, int n_in,
                              void* d_out, int out_size, void* d_ws, size_t ws_size,
                              hipStream_t stream) {
  (void)n_in; (void)out_size; (void)d_ws; (void)ws_size;
  const float* H  = (const float*)d_in[0];
  const float* y  = (const float*)d_in[1];
  const float* x  = (const float*)d_in[2];
  const float* ns = (const float*)d_in[3];
  const float* th = (const float*)d_in[4];
  const float* gm = (const float*)d_in[5];
  const float* cs = (const float*)d_in[6];
  float* out = (float*)d_out;
  const int B = in_sizes[3];   // noise_sigma has one element per batch
  hipLaunchKernelGGL(oampnet_cdna5_kernel, dim3(B), dim3(256), 0, stream,
                     H, y, x, ns, th, gm, cs, out);
}